// DistillationLossWithTopK_5858335392432
// MI455X (gfx1250) — compile-verified
//
#include <hip/hip_runtime.h>

// ---------------------------------------------------------------------------
// DistillationLossWithTopK for MI455X (gfx1250).
// B=2, S=1024, V=32000, K=128, T=5, alpha=0.7, pad_id=0.
//
// Bandwidth-bound: 524 MB of logits -> ~22.5us floor at 23.3 TB/s.
// One workgroup (16 wave32) per row:
//   - teacher row staged to LDS with CDNA5 global_load_async_to_lds_b128,
//     waited with the split ASYNCcnt counter (s_wait_asynccnt),
//   - student online double-temperature logsumexp overlaps the DMA,
//     streamed with b128 loads,
//   - exact 128th-largest teacher value via 3-round LDS radix select
//     (12+12+8 bits) with b128 LDS reads,
//   - collapsed KL: kl_row = B/A - thr/T - log A + LSE(s/T),
//     e_k = exp((t_k-thr)/T), A = sum e_k, B = sum e_k*(t_k-s_k)/T.
// LDS: 128000 (teacher) + 16384 (hist) + 256 -> ~145KB => 2 WGs/WGP.
// ---------------------------------------------------------------------------

#define B_      2
#define S_      1024
#define V_      32000
#define N_      (B_ * (S_ - 1))
#define K_      128u
#define TEMP_   5.0f
#define INV_T   (1.0f / TEMP_)
#define ALPHA_  0.7f
#define THREADS 512
#define NWAVES  (THREADS / 32)

// Monotone float -> uint key (descending top-K == largest keys).
__device__ __forceinline__ unsigned f2key(float f) {
  unsigned u = __float_as_uint(f);
  return (u & 0x80000000u) ? ~u : (u | 0x80000000u);
}
__device__ __forceinline__ float key2f(unsigned k) {
  unsigned u = (k & 0x80000000u) ? (k & 0x7fffffffu) : ~k;
  return __uint_as_float(u);
}

// Scan histogram from the top bin down; select bin holding the krem-th
// largest element. Returns selected digit and the residual count within it.
__device__ __forceinline__ void scan_select(const unsigned* hist, int nbins,
                                            unsigned krem, unsigned* sel,
                                            unsigned* new_krem) {
  unsigned cum = 0;
  int d = nbins - 1;
  for (; d > 0; --d) {
    unsigned c = hist[d];
    if (cum + c >= krem) break;
    cum += c;
  }
  *sel = (unsigned)d;
  *new_krem = krem - cum;
}

__global__ __launch_bounds__(THREADS)
void distill_row_kernel(const float* __restrict__ sL,
                        const float* __restrict__ tL,
                        const int* __restrict__ labels,
                        const int* __restrict__ attn,
                        float* __restrict__ accum) {
  __shared__ __align__(16) float t_lds[V_];   // 128000 B, teacher row
  __shared__ unsigned hist[4096];             // 16384 B, radix histograms
  __shared__ float red[64];                   // cross-wave reduction scratch
  __shared__ unsigned sh_sel, sh_krem, sh_thrkey, sh_ties, tie_ctr;
  __shared__ float sh_lse1, sh_lseT;

  const int row = blockIdx.x;                 // 0..N_-1
  const int b   = row / (S_ - 1);
  const int pos = row - b * (S_ - 1);
  const float* __restrict__ srow = sL + (size_t)(b * S_ + pos) * V_;
  const float* __restrict__ trow = tL + (size_t)(b * S_ + pos) * V_;
  const int tid = threadIdx.x;
  const int wid = tid >> 5;

  // ---- Phase 0: CDNA5 async DMA of the teacher row into LDS -------------
  {
    unsigned lbase = (unsigned)(size_t)&t_lds[0];   // low 32b of generic ptr
    for (int i = tid; i < V_ / 4; i += THREADS) {
      unsigned laddr = lbase + (unsigned)i * 16u;
      const float* g = trow + i * 4;
      asm volatile("global_load_async_to_lds_b128 %0, %1, off"
                   :: "v"(laddr), "v"(g)
                   : "memory");
    }
  }

  // ---- Phase 1 (overlaps DMA): student online logsumexp at T=1, T=5 ------
  // b128 streaming loads; per-vector max test, 4 exps per temperature.
  float m  = -3.402823466e38f;
  float s1 = 0.0f, sT = 0.0f;
  {
    const float4* __restrict__ s4 = (const float4*)srow;
    for (int i = tid; i < V_ / 4; i += THREADS) {
      float4 x = s4[i];
      float vmax = fmaxf(fmaxf(x.x, x.y), fmaxf(x.z, x.w));
      if (vmax > m) {
        float d = m - vmax;                   // <= 0
        s1 *= __expf(d);
        sT *= __expf(d * INV_T);
        m = vmax;
      }
      float d0 = x.x - m, d1 = x.y - m, d2 = x.z - m, d3 = x.w - m;
      s1 += __expf(d0) + __expf(d1) + __expf(d2) + __expf(d3);
      sT += __expf(d0 * INV_T) + __expf(d1 * INV_T) +
            __expf(d2 * INV_T) + __expf(d3 * INV_T);
    }
  }
  // wave32 reduction of (m, s1, sT)
  for (int off = 16; off > 0; off >>= 1) {
    float m2 = __shfl_xor(m, off, 32);
    float a2 = __shfl_xor(s1, off, 32);
    float b2 = __shfl_xor(sT, off, 32);
    float M  = fmaxf(m, m2);
    s1 = s1 * __expf(m - M) + a2 * __expf(m2 - M);
    sT = sT * __expf((m - M) * INV_T) + b2 * __expf((m2 - M) * INV_T);
    m = M;
  }
  if ((tid & 31) == 0) { red[wid] = m; red[16 + wid] = s1; red[32 + wid] = sT; }
  __syncthreads();
  if (tid == 0) {
    float M = red[0], S1 = red[16], ST = red[32];
    for (int w = 1; w < NWAVES; ++w) {
      float m2 = red[w], a2 = red[16 + w], b2 = red[32 + w];
      float Mn = fmaxf(M, m2);
      S1 = S1 * __expf(M - Mn) + a2 * __expf(m2 - Mn);
      ST = ST * __expf((M - Mn) * INV_T) + b2 * __expf((m2 - Mn) * INV_T);
      M = Mn;
    }
    sh_lse1 = M + __logf(S1);                // logsumexp(s)
    sh_lseT = M * INV_T + __logf(ST);        // logsumexp(s/T)
  }

  // ---- DMA completion: split-counter wait, then WG barrier ---------------
  asm volatile("s_wait_asynccnt 0" ::: "memory");
  __syncthreads();

  const float4* __restrict__ t4 = (const float4*)t_lds;

  // ---- Phase 2: exact 128th-largest teacher key via LDS radix select -----
  // Round 0: top 12 bits (b128 LDS reads, 4 ds_add per vector).
  for (int i = tid; i < 4096; i += THREADS) hist[i] = 0;
  __syncthreads();
  for (int i = tid; i < V_ / 4; i += THREADS) {
    float4 tv = t4[i];
    atomicAdd(&hist[f2key(tv.x) >> 20], 1u);
    atomicAdd(&hist[f2key(tv.y) >> 20], 1u);
    atomicAdd(&hist[f2key(tv.z) >> 20], 1u);
    atomicAdd(&hist[f2key(tv.w) >> 20], 1u);
  }
  __syncthreads();
  if (tid == 0) {
    unsigned sel, nk;
    scan_select(hist, 4096, K_, &sel, &nk);
    sh_sel = sel; sh_krem = nk;
  }
  __syncthreads();
  const unsigned p0 = sh_sel;
  const unsigned kr0 = sh_krem;

  // Round 1: next 12 bits among matching prefix.
  for (int i = tid; i < 4096; i += THREADS) hist[i] = 0;
  __syncthreads();
  for (int i = tid; i < V_ / 4; i += THREADS) {
    float4 tv = t4[i];
    unsigned k;
    k = f2key(tv.x); if ((k >> 20) == p0) atomicAdd(&hist[(k >> 8) & 0xFFFu], 1u);
    k = f2key(tv.y); if ((k >> 20) == p0) atomicAdd(&hist[(k >> 8) & 0xFFFu], 1u);
    k = f2key(tv.z); if ((k >> 20) == p0) atomicAdd(&hist[(k >> 8) & 0xFFFu], 1u);
    k = f2key(tv.w); if ((k >> 20) == p0) atomicAdd(&hist[(k >> 8) & 0xFFFu], 1u);
  }
  __syncthreads();
  if (tid == 0) {
    unsigned sel, nk;
    scan_select(hist, 4096, kr0, &sel, &nk);
    sh_sel = (p0 << 12) | sel; sh_krem = nk;
  }
  __syncthreads();
  const unsigned p1 = sh_sel;                // top 24 bits of threshold key
  const unsigned kr1 = sh_krem;

  // Round 2: low 8 bits.
  for (int i = tid; i < 256; i += THREADS) hist[i] = 0;
  __syncthreads();
  for (int i = tid; i < V_ / 4; i += THREADS) {
    float4 tv = t4[i];
    unsigned k;
    k = f2key(tv.x); if ((k >> 8) == p1) atomicAdd(&hist[k & 0xFFu], 1u);
    k = f2key(tv.y); if ((k >> 8) == p1) atomicAdd(&hist[k & 0xFFu], 1u);
    k = f2key(tv.z); if ((k >> 8) == p1) atomicAdd(&hist[k & 0xFFu], 1u);
    k = f2key(tv.w); if ((k >> 8) == p1) atomicAdd(&hist[k & 0xFFu], 1u);
  }
  __syncthreads();
  if (tid == 0) {
    unsigned sel, nk;
    scan_select(hist, 256, kr1, &sel, &nk);
    sh_thrkey = (p1 << 8) | sel;
    sh_ties = nk;                            // # threshold-equal elems to take
    tie_ctr = 0;
  }
  __syncthreads();

  // ---- Phase 3: collapsed KL accumulation over selected top-K ------------
  const unsigned thr_key = sh_thrkey;
  const unsigned ties    = sh_ties;
  const float thr = key2f(thr_key);
  float locA = 0.0f, locB = 0.0f;
  for (int i = tid; i < V_ / 4; i += THREADS) {
    float4 tvv = t4[i];
    float tv[4] = {tvv.x, tvv.y, tvv.z, tvv.w};
#pragma unroll
    for (int j = 0; j < 4; ++j) {
      unsigned k = f2key(tv[j]);
      bool take = (k > thr_key);
      if (k == thr_key) {
        unsigned slot = atomicAdd(&tie_ctr, 1u);
        take = (slot < ties);
      }
      if (take) {
        float e  = __expf((tv[j] - thr) * INV_T); // exp((t_k - thr)/T)
        float sv = srow[4 * i + j];               // L2-resident gather
        locA += e;
        locB += e * ((tv[j] - sv) * INV_T);
      }
    }
  }
  for (int off = 16; off > 0; off >>= 1) {
    locA += __shfl_xor(locA, off, 32);
    locB += __shfl_xor(locB, off, 32);
  }
  __syncthreads();                               // red[] reuse
  if ((tid & 31) == 0) { red[wid] = locA; red[16 + wid] = locB; }
  __syncthreads();

  if (tid == 0) {
    float A = 0.0f, Bs = 0.0f;
    for (int w = 0; w < NWAVES; ++w) { A += red[w]; Bs += red[16 + w]; }
    // kl_row = sum_k p_t*(t_k - s_k)/T - LSE_t_topk + LSE_s_full
    float kl_row = Bs / A + sh_lseT - thr * INV_T - __logf(A);

    int   lab   = labels[b * S_ + pos + 1];
    float maskv = (float)attn[b * S_ + pos + 1];
    float nll   = sh_lse1 - srow[lab];
    float valid = (lab != 0) ? 1.0f : 0.0f;

    atomicAdd(&accum[0], kl_row * maskv);
    atomicAdd(&accum[1], maskv);
    atomicAdd(&accum[2], nll * valid);
    atomicAdd(&accum[3], valid);
  }
}

__global__ void finalize_kernel(const float* __restrict__ accum,
                                float* __restrict__ out) {
  float kl = (accum[0] / accum[1]) * (TEMP_ * TEMP_);
  float ce = accum[2] / accum[3];
  out[0] = ALPHA_ * kl + (1.0f - ALPHA_) * ce;
}

extern "C" void kernel_launch(void* const* d_in, const int* in_sizes, int n_in,
                              void* d_out, int out_size, void* d_ws,
                              size_t ws_size, hipStream_t stream) {
  const float* sL     = (const float*)d_in[0];   // student_logits [B,S,V] f32
  const float* tL     = (const float*)d_in[1];   // teacher_logits [B,S,V] f32
  const int*   labels = (const int*)d_in[2];     // labels [B,S] int
  const int*   attn   = (const int*)d_in[3];     // attention_mask [B,S] int
  float* accum = (float*)d_ws;                   // 4 f32 accumulators

  hipMemsetAsync(d_ws, 0, 4 * sizeof(float), stream);
  distill_row_kernel<<<N_, THREADS, 0, stream>>>(sL, tL, labels, attn, accum);
  finalize_kernel<<<1, 1, 0, stream>>>(accum, (float*)d_out);
}